// CFConv_1623497638322
// MI455X (gfx1250) — compile-verified
//
#include <hip/hip_runtime.h>
#include <stdint.h>

#define N_FEAT 64

constexpr int kChunk         = 128;  // edges per wave
constexpr int kTile          = 8;    // edges per async double-buffer tile
constexpr int kWavesPerBlock = 8;
constexpr int kBlock         = 32 * kWavesPerBlock;

// gfx1250 async global->LDS path (ASYNCcnt). Fallback = direct global loads.
#if defined(__AMDGCN__) && defined(__gfx1250__) &&                         \
    __has_builtin(__builtin_amdgcn_global_load_async_to_lds_b64) &&        \
    __has_builtin(__builtin_amdgcn_s_wait_asynccnt)
#define CF_ASYNC 1
// Builtin signature (from clang diagnostic): first param is
//   int __attribute__((vector_size(8))) __device__ *   (AS1, int2 pointee)
typedef int v2i __attribute__((ext_vector_type(2)));
typedef __attribute__((address_space(1))) v2i glob_v2i;
typedef __attribute__((address_space(3))) v2i lds_v2i;
#define CF_COMPILER_FENCE() asm volatile("" ::: "memory")
#else
#define CF_ASYNC 0
#endif

__global__ void __launch_bounds__(kBlock) cfconv_zero(float* __restrict__ y, int n) {
  int i = blockIdx.x * blockDim.x + threadIdx.x;
  const int stride = gridDim.x * blockDim.x;
  for (; i < n; i += stride) y[i] = 0.0f;
}

__global__ void __launch_bounds__(kBlock) cfconv_kernel(
    const float* __restrict__ x, const float* __restrict__ Wij,
    const int* __restrict__ idx_i, const int* __restrict__ idx_j,
    float* __restrict__ y, int E) {
  __shared__ int lds_ii[kWavesPerBlock][kChunk];
  __shared__ int lds_jj[kWavesPerBlock][kChunk];
#if CF_ASYNC
  __shared__ float lds_w[kWavesPerBlock][2][kTile * N_FEAT];
#endif

  const int lane  = threadIdx.x & 31;          // wave32 (gfx1250)
  const int w     = threadIdx.x >> 5;
  const int gw    = blockIdx.x * kWavesPerBlock + w;
  const int start = gw * kChunk;
  if (start >= E) return;
  const int end = min(start + kChunk, E);
  const int cnt = end - start;

  // Stage this wave's edge indices into LDS (coalesced, then LDS-broadcast reads).
  for (int t = lane; t < cnt; t += 32) {
    lds_ii[w][t] = idx_i[start + t];
    lds_jj[w][t] = idx_j[start + t];
  }
  // Wave-private LDS: DS ops from one wave are in-order; no barrier needed.

  const int ntiles = (cnt + kTile - 1) / kTile;

#if CF_ASYNC
  auto prefetch = [&](int tileIdx, int par) {
    const int base = start + tileIdx * kTile;
#pragma unroll
    for (int t = 0; t < kTile; ++t) {
      int e = base + t;
      if (e >= E) e = E - 1;  // clamp: junk rows are never consumed
      const float* g = Wij + (size_t)e * N_FEAT + lane * 2;
      float*       l = &lds_w[w][par][t * N_FEAT + lane * 2];
      // per-lane 8B copy: 32 lanes x 8B = one 256B row per instruction
      __builtin_amdgcn_global_load_async_to_lds_b64((glob_v2i*)g, (lds_v2i*)l, 0, 0);
    }
    CF_COMPILER_FENCE();  // keep async issue ordered vs surrounding LDS traffic
  };
  prefetch(0, 0);
#endif

  float ax = 0.0f, ay = 0.0f;
  int cur = __builtin_amdgcn_readfirstlane(lds_ii[w][0]);
  int par = 0;

  for (int tile = 0; tile < ntiles; ++tile) {
    const bool more = (tile + 1) < ntiles;
#if CF_ASYNC
    if (more) {
      prefetch(tile + 1, par ^ 1);                 // issue next tile
      __builtin_amdgcn_s_wait_asynccnt(kTile);     // current tile done (in-order)
    } else {
      __builtin_amdgcn_s_wait_asynccnt(0);
    }
    CF_COMPILER_FENCE();  // consumer ds_loads must not hoist above the wait
#endif
    const int tbase = tile * kTile;
#pragma unroll
    for (int t = 0; t < kTile; ++t) {
      const int local = tbase + t;
      if (local >= cnt) break;
      const int seg = __builtin_amdgcn_readfirstlane(lds_ii[w][local]);
      const int j   = __builtin_amdgcn_readfirstlane(lds_jj[w][local]);
      float wx, wy;
#if CF_ASYNC
      wx = lds_w[w][par][t * N_FEAT + lane * 2 + 0];
      wy = lds_w[w][par][t * N_FEAT + lane * 2 + 1];
#else
      const float2 wv = *(const float2*)&Wij[(size_t)(start + local) * N_FEAT + lane * 2];
      wx = wv.x; wy = wv.y;
#endif
      // x gather: 12.8MB table, L2-resident on 192MB L2
      const float2 xj = *(const float2*)&x[(size_t)j * N_FEAT + lane * 2];

      if (seg != cur) {  // idx_i sorted -> flush only at segment boundaries
        float* yp = y + (size_t)cur * N_FEAT + lane * 2;
        unsafeAtomicAdd(yp + 0, ax);
        unsafeAtomicAdd(yp + 1, ay);
        ax = 0.0f; ay = 0.0f;
        cur = seg;
      }
      ax = fmaf(wx, xj.x, ax);
      ay = fmaf(wy, xj.y, ay);
    }
    par ^= 1;
  }

  {  // final flush (segment may continue into next wave's chunk -> atomics)
    float* yp = y + (size_t)cur * N_FEAT + lane * 2;
    unsafeAtomicAdd(yp + 0, ax);
    unsafeAtomicAdd(yp + 1, ay);
  }
}

extern "C" void kernel_launch(void* const* d_in, const int* in_sizes, int n_in,
                              void* d_out, int out_size, void* d_ws, size_t ws_size,
                              hipStream_t stream) {
  const float* x    = (const float*)d_in[0];
  const float* Wij  = (const float*)d_in[1];
  const int* idx_i  = (const int*)d_in[2];
  const int* idx_j  = (const int*)d_in[3];
  float* y          = (float*)d_out;
  const int E       = in_sizes[2];

  int zb = (out_size + kBlock - 1) / kBlock;
  if (zb > 1024) zb = 1024;
  cfconv_zero<<<zb, kBlock, 0, stream>>>(y, out_size);

  const int waves  = (E + kChunk - 1) / kChunk;
  const int blocks = (waves + kWavesPerBlock - 1) / kWavesPerBlock;
  cfconv_kernel<<<blocks, kBlock, 0, stream>>>(x, Wij, idx_i, idx_j, y, E);
}